// _RefMoEBlock_25159918420619
// MI455X (gfx1250) — compile-verified
//
#include <hip/hip_runtime.h>
#include <math.h>

// Problem constants (match reference)
#define T_TOK 8192
#define H_DIM 2048
#define I_DIM 1024
#define E_NUM 8
#define K_TOP 2
#define NENT  (T_TOK * K_TOP)   // 16384 routed (token, k) entries

// Workspace layout in 4-byte units
#define WS_TILEBASE 0                 // int tileBase[E_NUM+1]
#define WS_ENTBASE  16                // int entBase[E_NUM+1]
#define WS_TOK      32                // int tok[NENT]
#define WS_WGT      (32 + NENT)       // float wgt[NENT]

typedef float v2f __attribute__((ext_vector_type(2)));
typedef float v8f __attribute__((ext_vector_type(8)));

#define WAVES   8
#define BLOCK   (WAVES * 32)
#define CHUNK   256                   // K-chunk of hidden staged in LDS
#define A_STR   (CHUNK + 4)           // pad: row stride 260 -> bank spread, 16B aligned
#define ACT_STR (I_DIM + 4)           // pad: row stride 1028

static __device__ __forceinline__ v8f zero8() {
    v8f z = {0.f, 0.f, 0.f, 0.f, 0.f, 0.f, 0.f, 0.f};
    return z;
}

// -------------------------------------------------------------------------
// Kernel 1: deterministic routing / compaction.
// One block of 256 threads; each thread owns a contiguous slice of entries,
// so per-expert ordering is fixed -> fully deterministic placement.
// -------------------------------------------------------------------------
__global__ __launch_bounds__(256) void moe_route(
    const float* topw, const int* topi, const float* scal,
    int* wsi, float* wsf)
{
    __shared__ int lcnt[256][E_NUM];
    __shared__ int tot[E_NUM];
    __shared__ int entBase_s[E_NUM + 1];
    const int tid = threadIdx.x;
    const int per = NENT / 256;   // 64 entries per thread

    #pragma unroll
    for (int e = 0; e < E_NUM; ++e) lcnt[tid][e] = 0;
    const int i0 = tid * per;
    for (int i = 0; i < per; ++i) lcnt[tid][topi[i0 + i]]++;
    __syncthreads();

    if (tid < E_NUM) {            // per-expert exclusive prefix over threads
        int run = 0;
        for (int t = 0; t < 256; ++t) {
            int v = lcnt[t][tid];
            lcnt[t][tid] = run;
            run += v;
        }
        tot[tid] = run;
    }
    __syncthreads();

    if (tid == 0) {               // cross-expert bases + tile bases
        int eb = 0, tb = 0;
        for (int e = 0; e < E_NUM; ++e) {
            entBase_s[e] = eb;
            wsi[WS_ENTBASE + e]  = eb;
            wsi[WS_TILEBASE + e] = tb;
            eb += tot[e];
            tb += (tot[e] + 15) >> 4;
        }
        entBase_s[E_NUM] = eb;
        wsi[WS_ENTBASE + E_NUM]  = eb;
        wsi[WS_TILEBASE + E_NUM] = tb;
    }
    __syncthreads();

    for (int i = 0; i < per; ++i) {
        int ent = i0 + i;
        int e   = topi[ent];
        int pos = entBase_s[e] + lcnt[tid][e]++;
        wsi[WS_TOK + pos] = ent / K_TOP;                       // token id
        wsf[WS_WGT + pos] = topw[ent] * scal[e];               // combine weight
    }
}

// -------------------------------------------------------------------------
// Kernel 2: zero the output (fused kernel scatter-accumulates with atomics)
// -------------------------------------------------------------------------
__global__ __launch_bounds__(256) void moe_zero_out(float* out, int n4) {
    int i = blockIdx.x * blockDim.x + threadIdx.x;
    if (i < n4) ((float4*)out)[i] = make_float4(0.f, 0.f, 0.f, 0.f);
}

// -------------------------------------------------------------------------
// Kernel 3: fused per-tile expert kernel.
// One block = 16 routed rows of one expert. 8 waves split the N dimension.
// gate_up GEMM (K=2048, f32 WMMA) -> GeGLU (tanh GELU) -> LDS act -> down GEMM (K=1024)
// -> weighted atomic scatter into out.
// -------------------------------------------------------------------------
__global__ __launch_bounds__(BLOCK, 1) void moe_expert_tile(
    const float* __restrict__ hs, const float* __restrict__ gup,
    const float* __restrict__ dwn, float* out,
    const int* wsi, const float* wsf)
{
    __shared__ float Ab[2][16 * A_STR];    // 33,280 B: hidden K-chunks, dbl buffered
    __shared__ float ACT[16 * ACT_STR];    // 65,792 B: GeGLU activations
    __shared__ int   trow[16];
    __shared__ float wrow[16];

    const int g = blockIdx.x;
    if (g >= wsi[WS_TILEBASE + E_NUM]) return;      // uniform exit
    int e = 0;
    while (g >= wsi[WS_TILEBASE + e + 1]) ++e;
    const int entLo = wsi[WS_ENTBASE + e];
    const int entHi = wsi[WS_ENTBASE + e + 1];
    const int row0  = entLo + (g - wsi[WS_TILEBASE + e]) * 16;

    const int tid = threadIdx.x;
    if (tid < 16) {
        int r = row0 + tid;
        if (r < entHi) { trow[tid] = wsi[WS_TOK + r]; wrow[tid] = wsf[WS_WGT + r]; }
        else           { trow[tid] = 0;               wrow[tid] = 0.0f; }       // pad row
    }
    __syncthreads();

    const int lane = tid & 31;
    const int wave = tid >> 5;
    const int lm   = lane & 15;     // A frag: row M; B frag / acc: col N
    const int kh   = lane >> 4;     // 0: K{0,1}, 1: K{2,3} (A/B); acc: M += 8*kh

    const float* gup_e = gup + (size_t)e * (2 * I_DIM) * H_DIM;

    // ---- Phase 1: gu = h @ gate_up[e].T, accumulators persist across K-chunks
    v8f accg[8], accu[8];
    #pragma unroll
    for (int jj = 0; jj < 8; ++jj) { accg[jj] = zero8(); accu[jj] = zero8(); }

    // stage chunk 0
    for (int v = tid; v < 16 * (CHUNK / 4); v += BLOCK) {
        int row = v / (CHUNK / 4), c4 = v % (CHUNK / 4);
        float4 d = ((const float4*)(hs + (size_t)trow[row] * H_DIM))[c4];
        *(float4*)&Ab[0][row * A_STR + c4 * 4] = d;
    }
    __syncthreads();

    const int NCH = H_DIM / CHUNK;  // 8
    for (int c = 0; c < NCH; ++c) {
        const int cur = c & 1;
        if (c + 1 < NCH) {          // prefetch next chunk into other buffer
            for (int v = tid; v < 16 * (CHUNK / 4); v += BLOCK) {
                int row = v / (CHUNK / 4), c4 = v % (CHUNK / 4);
                float4 d = ((const float4*)(hs + (size_t)trow[row] * H_DIM
                                            + (size_t)(c + 1) * CHUNK))[c4];
                *(float4*)&Ab[cur ^ 1][row * A_STR + c4 * 4] = d;
            }
        }
        const float* Ap = &Ab[cur][lm * A_STR + 2 * kh];
        #pragma unroll
        for (int jj = 0; jj < 8; ++jj) {
            const int n0 = (wave + jj * WAVES) * 16;
            const float* Bg = gup_e + (size_t)(n0 + lm) * H_DIM
                            + (size_t)c * CHUNK + 2 * kh;
            const float* Bu = Bg + (size_t)I_DIM * H_DIM;
            v8f ag = accg[jj], au = accu[jj];
            #pragma unroll 8
            for (int k = 0; k < CHUNK; k += 4) {
                v2f a  = *(const v2f*)(Ap + k);
                v2f bg = *(const v2f*)(Bg + k);
                v2f bu = *(const v2f*)(Bu + k);
                ag = __builtin_amdgcn_wmma_f32_16x16x4_f32(false, a, false, bg,
                                                           (short)0, ag, false, false);
                au = __builtin_amdgcn_wmma_f32_16x16x4_f32(false, a, false, bu,
                                                           (short)0, au, false, false);
            }
            accg[jj] = ag; accu[jj] = au;
        }
        __syncthreads();
    }

    // GeGLU (tanh GELU) epilogue -> ACT (acc layout: vgpr i, lane -> M = i + 8*kh, N = lm)
    #pragma unroll
    for (int jj = 0; jj < 8; ++jj) {
        const int n0 = (wave + jj * WAVES) * 16;
        #pragma unroll
        for (int i = 0; i < 8; ++i) {
            int   m = i + 8 * kh;
            float x = accg[jj][i];
            float u = accu[jj][i];
            float t = tanhf(0.7978845608028654f * (x + 0.044715f * x * x * x));
            ACT[m * ACT_STR + n0 + lm] = 0.5f * x * (1.0f + t) * u;
        }
    }
    __syncthreads();

    // ---- Phase 2: cur = act @ down[e].T, weighted atomic scatter
    const float* dwn_e = dwn + (size_t)e * H_DIM * I_DIM;
    for (int j = wave; j < H_DIM / 16; j += WAVES) {
        const int h0 = j * 16;
        const float* Bd = dwn_e + (size_t)(h0 + lm) * I_DIM + 2 * kh;
        const float* Ap = &ACT[lm * ACT_STR + 2 * kh];
        v8f acc = zero8();
        #pragma unroll 8
        for (int k = 0; k < I_DIM; k += 4) {
            v2f a = *(const v2f*)(Ap + k);
            v2f b = *(const v2f*)(Bd + k);
            acc = __builtin_amdgcn_wmma_f32_16x16x4_f32(false, a, false, b,
                                                        (short)0, acc, false, false);
        }
        #pragma unroll
        for (int i = 0; i < 8; ++i) {
            int   m = i + 8 * kh;
            float w = wrow[m];
            if (w != 0.0f) {
                unsafeAtomicAdd(&out[(size_t)trow[m] * H_DIM + h0 + lm], acc[i] * w);
            }
        }
    }
}

// -------------------------------------------------------------------------
extern "C" void kernel_launch(void* const* d_in, const int* in_sizes, int n_in,
                              void* d_out, int out_size, void* d_ws, size_t ws_size,
                              hipStream_t stream)
{
    const float* hs   = (const float*)d_in[0];   // hidden_states  [T,H]
    const float* topw = (const float*)d_in[1];   // top_k_weights  [T,K]
    const float* gup  = (const float*)d_in[2];   // gate_up_proj   [E,2I,H]
    const float* dwn  = (const float*)d_in[3];   // down_proj      [E,H,I]
    const float* scal = (const float*)d_in[4];   // per_expert_scale [E]
    const int*   topi = (const int*)d_in[5];     // top_k_index    [T,K]
    float* out = (float*)d_out;
    int*   wsi = (int*)d_ws;
    float* wsf = (float*)d_ws;

    const int n4 = (T_TOK * H_DIM) / 4;
    hipLaunchKernelGGL(moe_zero_out, dim3((n4 + 255) / 256), dim3(256), 0, stream,
                       out, n4);
    hipLaunchKernelGGL(moe_route, dim3(1), dim3(256), 0, stream,
                       topw, topi, scal, wsi, wsf);
    const int maxTiles = NENT / 16 + E_NUM;      // 1032 upper bound (data-dep counts)
    hipLaunchKernelGGL(moe_expert_tile, dim3(maxTiles), dim3(BLOCK), 0, stream,
                       hs, gup, dwn, out, wsi, wsf);
}